// DimeNetPlusPlus_3822520894063
// MI455X (gfx1250) — compile-verified
//
#include <hip/hip_runtime.h>

// ---------------- problem constants ----------------
constexpr int Nn  = 50000;   // nodes
constexpr int Ee  = 800000;  // edges
constexpr int Tt  = 800000;  // triplets
constexpr int Gg  = 64;      // graphs
constexpr int Hh  = 128;     // hidden
constexpr int NRr = 6;       // radial basis
constexpr int NBb = 4;       // interaction blocks
constexpr float CUTi = 0.2f; // 1/CUT

typedef __attribute__((ext_vector_type(2))) float v2f;
typedef __attribute__((ext_vector_type(8))) float v8f;

__device__ __forceinline__ float siluf(float v) { return v / (1.0f + __expf(-v)); }

__device__ __forceinline__ float envf(float xr) {
    float x2 = xr * xr, x4 = x2 * x2;
    float x5 = x4 * xr, x6 = x5 * xr, x7 = x6 * xr;
    return 1.0f / xr - 28.0f * x5 + 48.0f * x6 - 21.0f * x7;
}

// ---------------- CDNA5 async global->LDS copy (ASYNCcnt path) ----------------
// Per-lane: copies 16 bytes from a global address to an LDS address, tracked by ASYNCcnt.
__device__ __forceinline__ void async_ld_b128(unsigned lds_addr, const void* gaddr) {
    asm volatile("global_load_async_to_lds_b128 %0, %1, off"
                 :: "v"(lds_addr), "v"(gaddr)
                 : "memory");
}
__device__ __forceinline__ void wait_async0() {
    asm volatile("s_wait_asynccnt 0x0" ::: "memory");
}

// ---------------- elementwise / basis kernels ----------------
__global__ void k_zero(float* p, int n) {
    int i = blockIdx.x * blockDim.x + threadIdx.x;
    if (i < n) p[i] = 0.0f;
}

__global__ void k_rbf(const float* __restrict__ dist, const float* __restrict__ freq,
                      float* __restrict__ rbf, int E_) {
    int e = blockIdx.x * blockDim.x + threadIdx.x;
    if (e >= E_) return;
    float xr = dist[e] * CUTi;
    float env = envf(xr);
#pragma unroll
    for (int r = 0; r < NRr; ++r) rbf[e * NRr + r] = env * __sinf(freq[r] * xr);
}

__global__ void k_sbf(const float* __restrict__ dist_trip, const float* __restrict__ freq,
                      const float* __restrict__ angle, float* __restrict__ sbf, int T_) {
    int t = blockIdx.x * blockDim.x + threadIdx.x;
    if (t >= T_) return;
    float xr = dist_trip[t] * CUTi;
    float env = envf(xr);
    float rb[NRr];
#pragma unroll
    for (int r = 0; r < NRr; ++r) rb[r] = env * __sinf(freq[r] * xr);
    float a = angle[t];
    float sp[4] = {1.0f, a, 0.5f * (3.0f * a * a - 1.0f), 0.5f * (5.0f * a * a * a - 3.0f * a)};
#pragma unroll
    for (int s = 0; s < 4; ++s)
#pragma unroll
        for (int r = 0; r < NRr; ++r) sbf[(size_t)t * 24 + s * NRr + r] = rb[r] * sp[s];
}

__global__ void k_init_x(const int* __restrict__ z, const float* __restrict__ table,
                         float* __restrict__ x, int total) {
    int i = blockIdx.x * blockDim.x + threadIdx.x;
    if (i >= total) return;
    int node = i >> 7, c = i & 127;
    x[i] = table[z[node] * Hh + c];
}

// hid[m*8+k] = silu(sum_r feat[m*fdim+r] * w1[r*8+k])
__global__ void k_hidden8(const float* __restrict__ feat, int fdim,
                          const float* __restrict__ w1, float* __restrict__ hid, int M_) {
    int i = blockIdx.x * blockDim.x + threadIdx.x;
    if (i >= M_ * 8) return;
    int m = i >> 3, k = i & 7;
    float s = 0.0f;
    for (int r = 0; r < fdim; ++r) s += feat[(size_t)m * fdim + r] * w1[r * 8 + k];
    hid[i] = siluf(s);
}

// ---------------- shared WMMA tile core ----------------
// Bs2 is the paired-K LDS image of B: Bs2[p*128 + n] = (B[2p][n], B[2p+1][n]).
// A frag (32-bit A 16x4): lanes 0-15 K+0,K+1 ; lanes 16-31 K+2,K+3 -> hi = lane>>4.
// Each B fragment is one aligned 8-byte LDS load straight into a VGPR pair.
__device__ __forceinline__ void wmma_tile_128(const float* __restrict__ Arow,
                                              const float2* __restrict__ Bs2,
                                              int l16, int hi, v8f acc[8]) {
    const float2* Bp = Bs2 + l16;
    for (int kk = 0; kk < 32; ++kk) {
        const int k0 = kk * 4 + hi * 2;
        v2f a;
        a.x = Arow[k0];
        a.y = Arow[k0 + 1];
        const float2* Brow = Bp + (2 * kk + hi) * 128;
#pragma unroll
        for (int j = 0; j < 8; ++j) {
            float2 bb = Brow[j * 16];
            v2f bf;
            bf.x = bb.x;
            bf.y = bb.y;
            acc[j] = __builtin_amdgcn_wmma_f32_16x16x4_f32(false, a, false, bf,
                                                           (short)0, acc[j], false, false);
        }
    }
}

// stage B (128x128 row-major, global) into paired-K LDS layout
__device__ __forceinline__ void stage_B_paired(const float* __restrict__ B,
                                               float2* __restrict__ Bs2, int tid) {
    for (int i = tid; i < 64 * 128; i += 128) {
        int p = i >> 7, n = i & 127;
        Bs2[i] = make_float2(B[(2 * p) * 128 + n], B[(2 * p + 1) * 128 + n]);
    }
}

// ---------------- fp32 WMMA GEMM: C[Mx128] = act(A[Mx128] @ B[128x128] + bias) ----------------
__global__ void __launch_bounds__(128)
gemm_n128_wmma(const float* __restrict__ A, const float* __restrict__ B,
               const float* __restrict__ bias, float* __restrict__ C, int M, int act) {
    __shared__ __align__(16) float As[64 * 128];   // 32 KB, row-major A tile
    __shared__ __align__(16) float Bs[128 * 128];  // 64 KB, paired-K B image
    const int tid  = threadIdx.x;
    const int wave = tid >> 5;
    const int lane = tid & 31;
    const int row0 = blockIdx.x * 64;

    __builtin_prefetch(B, 0, 1);

    // async copy of the A tile straight into LDS (no VGPR round-trip);
    // OOB rows (last tile) are zeroed with plain ds stores.
    const float4* A4 = (const float4*)A;
    const unsigned asBase = (unsigned)(uintptr_t)&As[0];
    for (int i = tid; i < 64 * 32; i += 128) {
        int r  = i >> 5;
        int gr = row0 + r;
        if (gr < M) {
            async_ld_b128(asBase + i * 16, (const void*)(A4 + (size_t)gr * 32 + (i & 31)));
        } else {
            ((float4*)As)[i] = make_float4(0.f, 0.f, 0.f, 0.f);
        }
    }

    stage_B_paired(B, (float2*)Bs, tid);

    wait_async0();
    __syncthreads();

    const int l16 = lane & 15;
    const int hi  = lane >> 4;
    const float* Arow = &As[(wave * 16 + l16) * 128];

    v8f acc[8] = {};
    wmma_tile_128(Arow, (const float2*)Bs, l16, hi, acc);

    const int mbase = wave * 16 + hi * 8;  // C layout: lanes16-31 hold M=8..15
#pragma unroll
    for (int j = 0; j < 8; ++j) {
#pragma unroll
        for (int r = 0; r < 8; ++r) {
            int gm = row0 + mbase + r;
            if (gm < M) {
                int gc  = j * 16 + l16;
                float v = acc[j][r];
                if (bias) v += bias[gc];
                if (act) v = siluf(v);
                C[(size_t)gm * 128 + gc] = v;
            }
        }
    }
}

// ---------------- fused edge embedding: x[col] += silu(rbf_e@W1+b1)@W2 + b2 ----------------
__global__ void __launch_bounds__(128)
edge_embed_wmma(const float* __restrict__ rbf_e, const float* __restrict__ w1,
                const float* __restrict__ b1, const float* __restrict__ w2,
                const float* __restrict__ b2, const int* __restrict__ col,
                float* __restrict__ x, int E_) {
    __shared__ __align__(16) float Hs[64 * 128];   // 32 KB hidden tile
    __shared__ __align__(16) float Bs[128 * 128];  // 64 KB paired-K w2 image
    const int tid  = threadIdx.x;
    const int wave = tid >> 5;
    const int lane = tid & 31;
    const int e0   = blockIdx.x * 64;

    __builtin_prefetch(w2, 0, 1);

    stage_B_paired(w2, (float2*)Bs, tid);

    // tiny 6->128 MLP layer computed straight into LDS
    for (int i = tid; i < 64 * 128; i += 128) {
        int r = i >> 7, c = i & 127;
        int ge  = e0 + r;
        float h = 0.0f;
        if (ge < E_) {
            float s = b1[c];
#pragma unroll
            for (int q = 0; q < NRr; ++q) s += rbf_e[(size_t)ge * NRr + q] * w1[q * 128 + c];
            h = siluf(s);
        }
        Hs[i] = h;
    }
    __syncthreads();

    const int l16 = lane & 15;
    const int hi  = lane >> 4;
    const float* Arow = &Hs[(wave * 16 + l16) * 128];

    v8f acc[8] = {};
    wmma_tile_128(Arow, (const float2*)Bs, l16, hi, acc);

    const int mbase = wave * 16 + hi * 8;
#pragma unroll
    for (int j = 0; j < 8; ++j) {
#pragma unroll
        for (int r = 0; r < 8; ++r) {
            int ge = e0 + mbase + r;
            if (ge < E_) {
                int gc  = j * 16 + l16;
                float v = acc[j][r] + b2[gc];
                atomicAdd(&x[(size_t)col[ge] * 128 + gc], v);
            }
        }
    }
}

// ---------------- edge scatter kernels (memory-bound; fused elementwise) ----------------
// out_block edge pass: h[col] += x[row] * (rbf_e @ wr)
__global__ void __launch_bounds__(128)
k_outblock_edge(const float* __restrict__ rbf_e, const float* __restrict__ wr,
                const float* __restrict__ x, const int* __restrict__ row,
                const int* __restrict__ col, float* __restrict__ h, int E_) {
    __shared__ float rb[NRr];
    int e = blockIdx.x;
    if (e >= E_) return;
    int c = threadIdx.x;
    if (c < NRr) rb[c] = rbf_e[(size_t)e * NRr + c];
    __syncthreads();
    float emb = 0.0f;
#pragma unroll
    for (int r = 0; r < NRr; ++r) emb += rb[r] * wr[r * 128 + c];
    float v = x[(size_t)row[e] * 128 + c] * emb;
    atomicAdd(&h[(size_t)col[e] * 128 + c], v);
}

// interaction edge pass: x[col] += x_up[row] * (hid8 @ w2) + extra
__global__ void __launch_bounds__(128)
k_inter_edge(const float* __restrict__ hid8, const float* __restrict__ w2,
             const float* __restrict__ x_up, const float* __restrict__ extra,
             const int* __restrict__ row, const int* __restrict__ col,
             float* __restrict__ x, int E_) {
    __shared__ float hh[8];
    int e = blockIdx.x;
    if (e >= E_) return;
    int c = threadIdx.x;
    if (c < 8) hh[c] = hid8[(size_t)e * 8 + c];
    __syncthreads();
    float emb = 0.0f;
#pragma unroll
    for (int k = 0; k < 8; ++k) emb += hh[k] * w2[k * 128 + c];
    float v = x_up[(size_t)row[e] * 128 + c] * emb + extra[c];
    atomicAdd(&x[(size_t)col[e] * 128 + c], v);
}

// extra[c] += sum_t tmat[j_idx[t]][c] * (hid8[t] @ w2)[c]
__global__ void __launch_bounds__(256)
k_trip_extra(const float* __restrict__ hid8, const float* __restrict__ w2,
             const float* __restrict__ tmat, const int* __restrict__ trip,
             float* __restrict__ extra, int T_) {
    __shared__ float red[256];
    int c    = threadIdx.x & 127;
    int half = threadIdx.x >> 7;
    int tb   = blockIdx.x * 512;
    float acc = 0.0f;
    for (int tt = half; tt < 512; tt += 2) {
        int t = tb + tt;
        if (t >= T_) break;
        int j = trip[(size_t)t * 3 + 1];
        float emb = 0.0f;
#pragma unroll
        for (int k = 0; k < 8; ++k) emb += hid8[(size_t)t * 8 + k] * w2[k * 128 + c];
        acc += tmat[(size_t)j * 128 + c] * emb;
    }
    red[threadIdx.x] = acc;
    __syncthreads();
    if (half == 0) atomicAdd(&extra[c], red[c] + red[c + 128]);
}

// P[n] (+)= h2[n] @ w + b   (128x1 GEMV)
__global__ void k_out_final(const float* __restrict__ h2, const float* __restrict__ w,
                            const float* __restrict__ b, float* __restrict__ P,
                            int N_, int accum) {
    int n = blockIdx.x * blockDim.x + threadIdx.x;
    if (n >= N_) return;
    float s = b[0];
    const float* hr = &h2[(size_t)n * 128];
#pragma unroll 8
    for (int c = 0; c < 128; ++c) s += hr[c] * w[c];
    if (accum) P[n] += s; else P[n] = s;
}

__global__ void k_pool_sum(const float* __restrict__ P, const int* __restrict__ batch,
                           float* __restrict__ pooled, float* __restrict__ cnt, int N_) {
    int i = blockIdx.x * blockDim.x + threadIdx.x;
    if (i >= N_) return;
    int g = batch[i];
    atomicAdd(&pooled[g], P[i]);
    atomicAdd(&cnt[g], 1.0f);
}

__global__ void k_pool_div(const float* __restrict__ pooled, const float* __restrict__ cnt,
                           float* __restrict__ out, int G_) {
    int g = blockIdx.x * blockDim.x + threadIdx.x;
    if (g < G_) out[g] = pooled[g] / fmaxf(cnt[g], 1.0f);
}

// ---------------- host orchestration ----------------
static inline int ceil_div(int a, int b) { return (a + b - 1) / b; }

extern "C" void kernel_launch(void* const* d_in, const int* in_sizes, int n_in,
                              void* d_out, int out_size, void* d_ws, size_t ws_size,
                              hipStream_t stream) {
    const int*   z          = (const int*)d_in[0];
    const float* dist       = (const float*)d_in[1];
    const float* dist_trip  = (const float*)d_in[2];
    const float* angle      = (const float*)d_in[3];
    const int*   edge_index = (const int*)d_in[4];
    const int*   triplets   = (const int*)d_in[5];
    const int*   batch      = (const int*)d_in[6];
    const float* atom_table = (const float*)d_in[7];
    const float* rbf_freq   = (const float*)d_in[8];
    const float* sbf_freq   = (const float*)d_in[9];
    const float* emb_w1     = (const float*)d_in[10];
    const float* emb_b1     = (const float*)d_in[11];
    const float* emb_w2     = (const float*)d_in[12];
    const float* emb_b2     = (const float*)d_in[13];
    const float* iw_rbf1    = (const float*)d_in[14];
    const float* iw_rbf2    = (const float*)d_in[15];
    const float* iw_sbf1    = (const float*)d_in[16];
    const float* iw_sbf2    = (const float*)d_in[17];
    const float* iw_t1      = (const float*)d_in[18];
    const float* iw_t2      = (const float*)d_in[19];
    const float* ib_t2      = (const float*)d_in[20];
    const float* iw_up      = (const float*)d_in[21];
    const float* ib_up      = (const float*)d_in[22];
    const float* iw_down    = (const float*)d_in[23];
    const float* ib_down    = (const float*)d_in[24];
    const float* ow_rbf     = (const float*)d_in[25];
    const float* ow0        = (const float*)d_in[26];
    const float* ob0        = (const float*)d_in[27];
    const float* ow1        = (const float*)d_in[28];
    const float* ob1        = (const float*)d_in[29];
    const float* ow2        = (const float*)d_in[30];
    const float* ob2        = (const float*)d_in[31];
    float* out = (float*)d_out;

    const int* row = edge_index;        // edge_index[0]
    const int* col = edge_index + Ee;   // edge_index[1]

    // workspace carve-up (floats)
    float* ws = (float*)d_ws;
    size_t off = 0;
    auto alloc = [&](size_t n) { float* p = ws + off; off += n; return p; };
    float* rbf_e  = alloc((size_t)Ee * NRr);
    float* sbf_t  = alloc((size_t)Tt * 24);
    float* x      = alloc((size_t)Nn * Hh);
    float* x_up   = alloc((size_t)Nn * Hh);
    float* x_down = alloc((size_t)Nn * Hh);
    float* t_tmp  = alloc((size_t)Nn * Hh);
    float* t_mat  = alloc((size_t)Nn * Hh);
    float* hbuf   = alloc((size_t)Nn * Hh);
    float* h1     = alloc((size_t)Nn * Hh);
    float* h2     = alloc((size_t)Nn * Hh);
    float* hid8e  = alloc((size_t)Ee * 8);
    float* hid8t  = alloc((size_t)Tt * 8);
    float* P      = alloc((size_t)Nn);
    float* extra  = alloc(128);
    float* pooled = alloc(Gg);
    float* cnt    = alloc(Gg);

    auto zero = [&](float* p, int n) {
        k_zero<<<ceil_div(n, 256), 256, 0, stream>>>(p, n);
    };
    auto gemm = [&](const float* A, const float* B, const float* bias, float* C, int M, int act) {
        gemm_n128_wmma<<<ceil_div(M, 64), 128, 0, stream>>>(A, B, bias, C, M, act);
    };
    auto out_block = [&](int k, int accum) {
        zero(hbuf, Nn * Hh);
        k_outblock_edge<<<Ee, 128, 0, stream>>>(rbf_e, ow_rbf + (size_t)k * NRr * Hh,
                                                x, row, col, hbuf, Ee);
        gemm(hbuf, ow0 + (size_t)k * Hh * Hh, ob0 + (size_t)k * Hh, h1, Nn, 1);
        gemm(h1,   ow1 + (size_t)k * Hh * Hh, ob1 + (size_t)k * Hh, h2, Nn, 1);
        k_out_final<<<ceil_div(Nn, 256), 256, 0, stream>>>(h2, ow2 + (size_t)k * Hh,
                                                           ob2 + k, P, Nn, accum);
    };

    // basis functions + node init
    k_rbf<<<ceil_div(Ee, 256), 256, 0, stream>>>(dist, rbf_freq, rbf_e, Ee);
    k_sbf<<<ceil_div(Tt, 256), 256, 0, stream>>>(dist_trip, sbf_freq, angle, sbf_t, Tt);
    k_init_x<<<ceil_div(Nn * Hh, 256), 256, 0, stream>>>(z, atom_table, x, Nn * Hh);

    // fused edge embedding (WMMA) scatter-added into x
    edge_embed_wmma<<<ceil_div(Ee, 64), 128, 0, stream>>>(rbf_e, emb_w1, emb_b1, emb_w2,
                                                          emb_b2, col, x, Ee);

    // output block 0 (writes P)
    out_block(0, 0);

    // interaction blocks
    for (int b = 0; b < NBb; ++b) {
        gemm(x, iw_up   + (size_t)b * Hh * Hh, ib_up   + (size_t)b * Hh, x_up,   Nn, 0);
        gemm(x, iw_down + (size_t)b * Hh * Hh, ib_down + (size_t)b * Hh, x_down, Nn, 0);
        gemm(x_down, iw_t1 + (size_t)b * Hh * Hh, nullptr,                t_tmp, Nn, 1);
        gemm(t_tmp,  iw_t2 + (size_t)b * Hh * Hh, ib_t2 + (size_t)b * Hh, t_mat, Nn, 0);

        k_hidden8<<<ceil_div(Ee * 8, 256), 256, 0, stream>>>(rbf_e, NRr,
                                                             iw_rbf1 + (size_t)b * NRr * 8,
                                                             hid8e, Ee);
        k_hidden8<<<ceil_div(Tt * 8, 256), 256, 0, stream>>>(sbf_t, 24,
                                                             iw_sbf1 + (size_t)b * 24 * 8,
                                                             hid8t, Tt);
        zero(extra, 128);
        k_trip_extra<<<ceil_div(Tt, 512), 256, 0, stream>>>(hid8t,
                                                            iw_sbf2 + (size_t)b * 8 * Hh,
                                                            t_mat, triplets, extra, Tt);
        k_inter_edge<<<Ee, 128, 0, stream>>>(hid8e, iw_rbf2 + (size_t)b * 8 * Hh,
                                             x_up, extra, row, col, x, Ee);
        out_block(b + 1, 1);
    }

    // mean pool per graph
    zero(pooled, Gg);
    zero(cnt, Gg);
    k_pool_sum<<<ceil_div(Nn, 256), 256, 0, stream>>>(P, batch, pooled, cnt, Nn);
    k_pool_div<<<1, Gg, 0, stream>>>(pooled, cnt, out, Gg);
}